// GAT_82497731822208
// MI455X (gfx1250) — compile-verified
//
#include <hip/hip_runtime.h>
#include <hip/hip_bf16.h>
#include <math.h>

typedef float v2f __attribute__((ext_vector_type(2)));
typedef float v8f __attribute__((ext_vector_type(8)));

#define N_NODES 50000
#define N_EDGES 800000
#define IN_DIM  128
#define HID     64
#define HEADS   4
#define OUT_DIM 64
#define NEG_SLOPE 0.2f

static inline int cdiv(int a, int b) { return (a + b - 1) / b; }

// ---------------------------------------------------------------------------
// Generic fp32 WMMA GEMM: C[M,N] = A[M,K] @ B[K,N], row-major, K <= 256,
// K % 4 == 0.  Block = 256 threads = 8 waves; each block owns 16 rows of A
// (staged once into LDS) and 8 adjacent 16-wide N tiles (one per wave).
// Uses V_WMMA_F32_16X16X4_F32 (fp32 in/out — matches reference precision).
//
// ISA layouts (CDNA5 ISA §7.12.2):
//   A (16x4, 2 VGPRs/lane): lanes 0-15 -> (M=lane, K=k+0 / k+1)
//                           lanes16-31 -> (M=lane-16, K=k+2 / k+3)
//   B (4x16) row-striped across lanes, symmetric K split across lane halves.
//   C/D (16x16, 8 VGPRs): vgpr r, lanes 0-15 -> M=r, lanes 16-31 -> M=r+8.
// ---------------------------------------------------------------------------
__global__ __launch_bounds__(256) void gemm_f32_wmma(
    const float* __restrict__ A, const float* __restrict__ B,
    float* __restrict__ C, int M, int N, int K)
{
    __shared__ float As[16 * 256];

    const int tileM = blockIdx.y * 16;
    const int wave  = threadIdx.x >> 5;
    const int lane  = threadIdx.x & 31;
    const int tileN = (blockIdx.x * 8 + wave) * 16;

    // cooperative, coalesced stage of the 16xK A tile into LDS
    const int elems = 16 * K;
    for (int i = threadIdx.x; i < elems; i += 256) {
        const int r = i / K, c = i - r * K;
        const int gr = tileM + r;
        As[r * K + c] = (gr < M) ? A[(size_t)gr * K + c] : 0.0f;
    }
    __syncthreads();

    if (tileN >= N) return;

    v8f acc = {};
    const int half = lane >> 4;      // 0: lanes 0-15, 1: lanes 16-31
    const int l16  = lane & 15;

    for (int k = 0; k < K; k += 4) {
        v2f a;
        a.x = As[l16 * K + k + 2 * half + 0];
        a.y = As[l16 * K + k + 2 * half + 1];
        v2f b;
        const float* Bp = B + (size_t)(k + 2 * half) * N + (tileN + l16);
        b.x = Bp[0];
        b.y = Bp[N];
        acc = __builtin_amdgcn_wmma_f32_16x16x4_f32(
            false, a, false, b, (short)0, acc, false, false);
    }

    const int rowBase = tileM + 8 * half;
    const int col     = tileN + l16;
    if (col < N) {
        #pragma unroll
        for (int r = 0; r < 8; ++r) {
            const int row = rowBase + r;
            if (row < M) C[(size_t)row * N + col] = acc[r];
        }
    }
}

// ---------------------------------------------------------------------------
// el[n,h] = feat[n,h,:]·al[h,:]   er[n,h] = feat[n,h,:]·ar[h,:]
// ---------------------------------------------------------------------------
__global__ void attn_scores(const float* __restrict__ feat,
                            const float* __restrict__ al,
                            const float* __restrict__ ar,
                            float* __restrict__ el, float* __restrict__ er,
                            int n, int H, int D)
{
    const int i = blockIdx.x * blockDim.x + threadIdx.x;   // node*H + h
    if (i >= n * H) return;
    const int h = i % H;
    const float* f = feat + (size_t)(i / H) * H * D + h * D;
    const float* a = al + h * D;
    const float* b = ar + h * D;
    float sl = 0.f, sr = 0.f;
    for (int d = 0; d < D; d += 4) {
        const float4 fv = *(const float4*)(f + d);
        const float4 av = *(const float4*)(a + d);
        const float4 bv = *(const float4*)(b + d);
        sl += fv.x * av.x + fv.y * av.y + fv.z * av.z + fv.w * av.w;
        sr += fv.x * bv.x + fv.y * bv.y + fv.z * bv.z + fv.w * bv.w;
    }
    el[i] = sl;
    er[i] = sr;
}

__global__ void fill_u32(unsigned int* __restrict__ p, unsigned int v, int n)
{
    const int i = blockIdx.x * blockDim.x + threadIdx.x;
    if (i < n) p[i] = v;
}

__device__ inline float lrelu(float x) { return x > 0.f ? x : NEG_SLOPE * x; }

// float atomic max via int ordering trick (valid with -inf init)
__device__ inline void atomicMaxF(float* addr, float v)
{
    if (v >= 0.f) atomicMax((int*)addr, __float_as_int(v));
    else          atomicMin((unsigned int*)addr, __float_as_uint(v));
}

__global__ void edge_max(const int* __restrict__ src, const int* __restrict__ dst,
                         const float* __restrict__ el, const float* __restrict__ er,
                         float* __restrict__ emax, int E, int H)
{
    const int i = blockIdx.x * blockDim.x + threadIdx.x;   // edge*H + h
    if (i >= E * H) return;
    const int e = i / H, h = i - e * H;
    const float v = lrelu(el[src[e] * H + h] + er[dst[e] * H + h]);
    atomicMaxF(&emax[dst[e] * H + h], v);
}

__global__ void edge_scatter(const int* __restrict__ src, const int* __restrict__ dst,
                             const float* __restrict__ el, const float* __restrict__ er,
                             const float* __restrict__ emax,
                             const float* __restrict__ feat,
                             float* __restrict__ denom, float* __restrict__ numer,
                             int E, int H, int D)
{
    const int i = blockIdx.x * blockDim.x + threadIdx.x;   // edge*H + h
    if (i >= E * H) return;
    const int e = i / H, h = i - e * H;
    const int s = src[e], d = dst[e];
    const float ev = lrelu(el[s * H + h] + er[d * H + h]);
    const float es = expf(ev - emax[d * H + h]);
    atomicAdd(&denom[d * H + h], es);
    const float* f = feat  + (size_t)s * H * D + h * D;
    float*       o = numer + (size_t)d * H * D + h * D;
    for (int j = 0; j < D; j += 4) {
        const float4 fv = *(const float4*)(f + j);
        atomicAdd(o + j + 0, es * fv.x);
        atomicAdd(o + j + 1, es * fv.y);
        atomicAdd(o + j + 2, es * fv.z);
        atomicAdd(o + j + 3, es * fv.w);
    }
}

// out = ELU(numer/denom + bias)   (safe to run in place: out == numer)
__global__ void finalize_gat(const float* __restrict__ numer,
                             const float* __restrict__ denom,
                             const float* __restrict__ bias,
                             float* __restrict__ out, int n, int H, int D)
{
    const int i = blockIdx.x * blockDim.x + threadIdx.x;
    if (i >= n * H * D) return;
    const int hd   = i % (H * D);
    const int node = i / (H * D);
    const int h    = hd / D;
    const float den = fmaxf(denom[node * H + h], 1e-9f);
    const float v   = numer[i] / den + bias[hd];
    out[i] = v > 0.f ? v : expm1f(v);
}

// out = w * x2 + (1-w) * ELU(res_pre + res_b)
__global__ void fuse_residual(const float* __restrict__ x2,
                              const float* __restrict__ res_pre,
                              const float* __restrict__ res_b,
                              const float* __restrict__ fuse_w,
                              float* __restrict__ out, int n, int D)
{
    const int i = blockIdx.x * blockDim.x + threadIdx.x;
    if (i >= n * D) return;
    float r = res_pre[i] + res_b[i % D];
    r = r > 0.f ? r : expm1f(r);
    const float w = fuse_w[0];
    out[i] = w * x2[i] + (1.f - w) * r;
}

extern "C" void kernel_launch(void* const* d_in, const int* in_sizes, int n_in,
                              void* d_out, int out_size, void* d_ws, size_t ws_size,
                              hipStream_t stream)
{
    const float* features = (const float*)d_in[0];   // [N,128]
    const float* resfeat  = (const float*)d_in[1];   // [N,128]
    const int*   src      = (const int*)d_in[2];     // [E]
    const int*   dst      = (const int*)d_in[3];     // [E]
    const float* W1       = (const float*)d_in[4];   // [128,256]
    const float* al1      = (const float*)d_in[5];   // [4,64]
    const float* ar1      = (const float*)d_in[6];   // [4,64]
    const float* b1       = (const float*)d_in[7];   // [256]
    const float* W2       = (const float*)d_in[8];   // [256,64]
    const float* al2      = (const float*)d_in[9];   // [1,64]
    const float* ar2      = (const float*)d_in[10];  // [1,64]
    const float* b2       = (const float*)d_in[11];  // [64]
    const float* res_W    = (const float*)d_in[12];  // [128,64]
    const float* res_b    = (const float*)d_in[13];  // [64]
    const float* fuse_w   = (const float*)d_in[14];  // [1]

    const int N = N_NODES, E = N_EDGES;

    // workspace layout (fp32)
    float* ws     = (float*)d_ws;
    float* feat1  = ws;                  ws += (size_t)N * HEADS * HID;   // [N,256]
    float* numer1 = ws;                  ws += (size_t)N * HEADS * HID;   // [N,256] -> x1 (in place)
    float* el1    = ws;                  ws += (size_t)N * HEADS;
    float* er1    = ws;                  ws += (size_t)N * HEADS;
    float* emax1  = ws;                  ws += (size_t)N * HEADS;
    float* denom1 = ws;                  ws += (size_t)N * HEADS;
    float* feat2  = ws;                  ws += (size_t)N * HID;           // [N,64]
    float* numer2 = ws;                  ws += (size_t)N * HID;           // [N,64] -> x2 (in place)
    float* el2    = ws;                  ws += (size_t)N;
    float* er2    = ws;                  ws += (size_t)N;
    float* emax2  = ws;                  ws += (size_t)N;
    float* denom2 = ws;                  ws += (size_t)N;
    float* resp   = ws;                  ws += (size_t)N * OUT_DIM;       // [N,64]

    const int TB = 256;
    const unsigned NEG_INF = 0xFF800000u;

    // ---------------- layer 1 ----------------
    {
        dim3 g(cdiv(HEADS * HID, 128), cdiv(N, 16));
        gemm_f32_wmma<<<g, TB, 0, stream>>>(features, W1, feat1, N, HEADS * HID, IN_DIM);
    }
    attn_scores<<<cdiv(N * HEADS, TB), TB, 0, stream>>>(feat1, al1, ar1, el1, er1, N, HEADS, HID);

    fill_u32<<<cdiv(N * HEADS, TB), TB, 0, stream>>>((unsigned*)emax1, NEG_INF, N * HEADS);
    fill_u32<<<cdiv(N * HEADS, TB), TB, 0, stream>>>((unsigned*)denom1, 0u, N * HEADS);
    fill_u32<<<cdiv(N * HEADS * HID, TB), TB, 0, stream>>>((unsigned*)numer1, 0u, N * HEADS * HID);

    edge_max<<<cdiv(E * HEADS, TB), TB, 0, stream>>>(src, dst, el1, er1, emax1, E, HEADS);
    edge_scatter<<<cdiv(E * HEADS, TB), TB, 0, stream>>>(src, dst, el1, er1, emax1, feat1,
                                                         denom1, numer1, E, HEADS, HID);
    finalize_gat<<<cdiv(N * HEADS * HID, TB), TB, 0, stream>>>(numer1, denom1, b1, numer1,
                                                               N, HEADS, HID);  // numer1 == x1

    // ---------------- layer 2 ----------------
    {
        dim3 g(cdiv(HID, 128), cdiv(N, 16));
        gemm_f32_wmma<<<g, TB, 0, stream>>>(numer1, W2, feat2, N, HID, HEADS * HID);
    }
    attn_scores<<<cdiv(N, TB), TB, 0, stream>>>(feat2, al2, ar2, el2, er2, N, 1, HID);

    fill_u32<<<cdiv(N, TB), TB, 0, stream>>>((unsigned*)emax2, NEG_INF, N);
    fill_u32<<<cdiv(N, TB), TB, 0, stream>>>((unsigned*)denom2, 0u, N);
    fill_u32<<<cdiv(N * HID, TB), TB, 0, stream>>>((unsigned*)numer2, 0u, N * HID);

    edge_max<<<cdiv(E, TB), TB, 0, stream>>>(src, dst, el2, er2, emax2, E, 1);
    edge_scatter<<<cdiv(E, TB), TB, 0, stream>>>(src, dst, el2, er2, emax2, feat2,
                                                 denom2, numer2, E, 1, HID);
    finalize_gat<<<cdiv(N * HID, TB), TB, 0, stream>>>(numer2, denom2, b2, numer2,
                                                       N, 1, HID);              // numer2 == x2

    // ---------------- residual + fuse ----------------
    {
        dim3 g(cdiv(OUT_DIM, 128), cdiv(N, 16));
        gemm_f32_wmma<<<g, TB, 0, stream>>>(resfeat, res_W, resp, N, OUT_DIM, IN_DIM);
    }
    fuse_residual<<<cdiv(N * OUT_DIM, TB), TB, 0, stream>>>(numer2, resp, res_b, fuse_w,
                                                            (float*)d_out, N, OUT_DIM);
}